// PowerIteration_74010876444942
// MI455X (gfx1250) — compile-verified
//
#include <hip/hip_runtime.h>
#include <cstdint>
#include <cstddef>

// Power iteration, B=32, n=1024, fp32, 10 iterations, freeze-on-convergence.
// Bandwidth-bound (0.5 flop/byte): stream M at full rate, do the per-row dot
// products with V_WMMA_F32_16X16X4_F32 (exact fp32 math, waste-free under the
// memory roofline), stage M tiles LDS-side with double-buffered
// global_load_async_to_lds_b128 (ASYNCcnt-tracked CDNA5 async path).

#define NDIM 1024
#define BATCH 32
#define KT 64            // K-tile (floats) per async-staged LDS tile
#define TSTRIDE 68       // padded LDS row stride in floats (272B: 16B-aligned, conflict-free)
#define TOL2 1e-12f      // (1e-6)^2

typedef __attribute__((ext_vector_type(2))) float v2f;
typedef __attribute__((ext_vector_type(8))) float v8f;

// ---------------------------------------------------------------------------
// K1: vn_raw[b, rowBase..rowBase+15] = M[b] tile-block @ v[b]   (one wave = 16 rows)
// ---------------------------------------------------------------------------
__global__ __launch_bounds__(256) void pi_matvec_wmma(const float* __restrict__ M,
                                                      const float* __restrict__ v,
                                                      float* __restrict__ vn) {
  __shared__ __align__(16) float vsh[NDIM];                 // 4 KB: v[b] staged once
  __shared__ __align__(16) float tiles[8][2][16 * TSTRIDE]; // 8 waves x double buffer

  const int tid  = threadIdx.x;
  const int wv   = tid >> 5;     // wave within block (wave32)
  const int lane = tid & 31;
  const int b    = blockIdx.y;
  const int rowBase = blockIdx.x * 128 + wv * 16;

  for (int i = tid; i < NDIM; i += 256) vsh[i] = v[b * NDIM + i];
  __syncthreads();

  const int h   = lane >> 4;     // lane half: selects K-pair per ISA A/B layout
  const int l15 = lane & 15;

  // Async-copy addressing: issue i (0..7) moves rows {2i+h} of the 16-row tile,
  // 16B segment l15, i.e. seg = i*32 + lane covers the whole 16x64 fp32 tile.
  const unsigned long long gbase =
      (unsigned long long)(const void*)(M + ((size_t)b * NDIM + rowBase) * NDIM) +
      (unsigned long long)(h * (NDIM * 4) + l15 * 16);

  const unsigned ldsb0 = (unsigned)(uintptr_t)&tiles[wv][0][0] +
                         (unsigned)(h * (TSTRIDE * 4) + l15 * 16);
  const unsigned ldsb1 = (unsigned)(uintptr_t)&tiles[wv][1][0] +
                         (unsigned)(h * (TSTRIDE * 4) + l15 * 16);

  auto issue_tile = [&](unsigned lbase, int kBase) {
    const unsigned long long g = gbase + (unsigned long long)(kBase * 4);
#pragma unroll
    for (int i = 0; i < 8; ++i) {
      asm volatile("global_load_async_to_lds_b128 %0, %1, off"
                   :: "v"(lbase + (unsigned)(i * 2 * TSTRIDE * 4)),
                      "v"(g + (unsigned long long)(i * 2 * NDIM * 4))
                   : "memory");
    }
  };

  v8f acc;
#pragma unroll
  for (int r = 0; r < 8; ++r) acc[r] = 0.0f;

  issue_tile(ldsb0, 0);
  issue_tile(ldsb1, KT);

  // A operand pointer: row l15, columns (4c + 2h, +1) -> ds_load_b64 per chunk
  const float* tp0 = &tiles[wv][0][l15 * TSTRIDE + 2 * h];
  const float* tp1 = &tiles[wv][1][l15 * TSTRIDE + 2 * h];
  // B operand pointer: lane-half offset baked into the address so each chunk is
  // a single broadcast ds_load_b64 (no cndmask): bb = { v[k+2h], v[k+2h+1] }.
  const float* tpv = vsh + 2 * h;

#pragma unroll 1
  for (int t = 0; t < NDIM / KT; ++t) {
    if (t < NDIM / KT - 1)
      asm volatile("s_wait_asynccnt 0x8" ::: "memory"); // oldest tile (8 ops) done
    else
      asm volatile("s_wait_asynccnt 0x0" ::: "memory"); // last tile: drain

    const int buf = t & 1;
    const float* tp = buf ? tp1 : tp0;
    const int kBase = t * KT;

#pragma unroll
    for (int c = 0; c < KT / 4; ++c) {
      // A 16x4 fp32 (ISA layout): lane half picks K-pair, a.x->K, a.y->K+1.
      v2f a = *(const v2f*)(tp + 4 * c);
      // B 4x16, columns broadcast: every D column equals the matvec result.
      v2f bb = *(const v2f*)(tpv + kBase + 4 * c);
      acc = __builtin_amdgcn_wmma_f32_16x16x4_f32(false, a, false, bb,
                                                  (short)0, acc, false, false);
    }
    if (t + 2 < NDIM / KT) issue_tile(buf ? ldsb1 : ldsb0, (t + 2) * KT);
  }

  // D layout: lane 0 VGPR r = row r, lane 16 VGPR r = row r+8 (column 0 etc).
  if (l15 == 0) {
    float* out = vn + (size_t)b * NDIM + rowBase + h * 8;
#pragma unroll
    for (int r = 0; r < 8; ++r) out[r] = acc[r];
  }
}

// ---------------------------------------------------------------------------
// K2: per-batch L2 norm of vn_raw
// ---------------------------------------------------------------------------
__global__ __launch_bounds__(256) void pi_batch_norm(const float* __restrict__ vn,
                                                     float* __restrict__ norms) {
  __shared__ float red[256];
  const int b = blockIdx.x, tid = threadIdx.x;
  float s = 0.0f;
  for (int i = tid; i < NDIM; i += 256) {
    float x = vn[b * NDIM + i];
    s += x * x;
  }
  red[tid] = s;
  __syncthreads();
  for (int off = 128; off > 0; off >>= 1) {
    if (tid < off) red[tid] += red[tid + off];
    __syncthreads();
  }
  if (tid == 0) norms[b] = sqrtf(red[0]);
}

// ---------------------------------------------------------------------------
// K3: vn = vn_raw / norm_b; per-block partial sums of (vn - vpre)^2 (deterministic)
// ---------------------------------------------------------------------------
__global__ __launch_bounds__(256) void pi_normalize_diff(float* __restrict__ vn,
                                                         const float* __restrict__ norms,
                                                         const float* __restrict__ vpre,
                                                         float* __restrict__ partials) {
  __shared__ float red[256];
  const int tid = threadIdx.x;
  const int i = blockIdx.x * 256 + tid;
  const int b = i >> 10;
  const float x = vn[i] / norms[b];
  vn[i] = x;
  const float d = x - vpre[i];
  red[tid] = d * d;
  __syncthreads();
  for (int off = 128; off > 0; off >>= 1) {
    if (tid < off) red[tid] += red[tid + off];
    __syncthreads();
  }
  if (tid == 0) partials[blockIdx.x] = red[0];
}

// ---------------------------------------------------------------------------
// K4: freeze-aware state update; thread 0 folds partials serially (deterministic)
// ---------------------------------------------------------------------------
__global__ __launch_bounds__(256) void pi_update(float* __restrict__ v,
                                                 float* __restrict__ vpre,
                                                 const float* __restrict__ vn,
                                                 const float* __restrict__ partials,
                                                 const int* __restrict__ done_in,
                                                 int* __restrict__ done_out) {
  const int i = blockIdx.x * 256 + threadIdx.x;
  const int d = done_in[0];
  if (!d) {                       // v_out = where(done_old, v, vn); same for vpre
    const float x = vn[i];
    v[i] = x;
    vpre[i] = x;
  }
  if (i == 0) {
    float s = 0.0f;
    for (int j = 0; j < 128; ++j) s += partials[j];
    done_out[0] = d | (s < TOL2 ? 1 : 0);
  }
}

// ---------------------------------------------------------------------------
// K0: v = vpre = 1, done = 0
// ---------------------------------------------------------------------------
__global__ __launch_bounds__(256) void pi_init(float* __restrict__ v,
                                               float* __restrict__ vpre,
                                               int* __restrict__ done0) {
  const int i = blockIdx.x * 256 + threadIdx.x;
  v[i] = 1.0f;
  vpre[i] = 1.0f;
  if (i == 0) done0[0] = 0;
}

extern "C" void kernel_launch(void* const* d_in, const int* in_sizes, int n_in,
                              void* d_out, int out_size, void* d_ws, size_t ws_size,
                              hipStream_t stream) {
  const float* M = (const float*)d_in[0];   // [32, 1024, 1024] fp32

  float* ws = (float*)d_ws;
  float* v        = ws;                     // 32768
  float* vpre     = ws + 32768;             // 32768
  float* vn       = ws + 65536;             // 32768
  float* norms    = ws + 98304;             // 32
  float* partials = ws + 98336;             // 128
  int*   done     = (int*)(ws + 98464);     // 2 (ping-pong)

  pi_init<<<128, 256, 0, stream>>>(v, vpre, done);

  for (int it = 0; it < 10; ++it) {
    pi_matvec_wmma<<<dim3(8, BATCH), 256, 0, stream>>>(M, v, vn);
    pi_batch_norm<<<BATCH, 256, 0, stream>>>(vn, norms);
    pi_normalize_diff<<<128, 256, 0, stream>>>(vn, norms, vpre, partials);
    pi_update<<<128, 256, 0, stream>>>(v, vpre, vn, partials,
                                       done + (it & 1), done + ((it + 1) & 1));
  }

  hipMemcpyAsync(d_out, v, (size_t)BATCH * NDIM * sizeof(float),
                 hipMemcpyDeviceToDevice, stream);
}